// PDFSampler_14542759264888
// MI455X (gfx1250) — compile-verified
//
#include <hip/hip_runtime.h>

// PDF inverse-transform sampler (NeRF fine-sample resampling), MI455X/gfx1250.
//
// Design (see analysis): memory-bound (~336 MB @ 23.3 TB/s ≈ 14.5 us floor).
// wave32: one wave per ray; 4 elements/lane. Single cross-lane scan of
// delta*density; CDF via telescoped exp (cumsum(weights) == 1 - exp(-incl)).
// CDNA5-specific: bins row streamed global->LDS with
// global_load_async_to_lds_b32 (ASYNCcnt), waited with s_wait_asynccnt 0
// right before the LDS binary search, fully overlapped with CDF construction.

#define NC  128
#define NS  128
#define WPB 8          // waves (rays) per block
#define EPSF 1e-5f

__launch_bounds__(256)
__global__ void pdf_sampler_kernel(const float* __restrict__ deltas,
                                   const float* __restrict__ density,
                                   const float* __restrict__ bins,
                                   const float* __restrict__ u,
                                   float* __restrict__ out,
                                   int R)
{
    // 132-wide rows: 129 live entries + sentinel/pad (bank-conflict slack)
    __shared__ float s_bins[WPB][132];
    __shared__ float s_cdf [WPB][132];

    const int w    = threadIdx.x >> 5;   // wave id in block
    const int lane = threadIdx.x & 31;   // wave32 lane
    const int r    = blockIdx.x * WPB + w;
    if (r >= R) return;

    // ---------------------------------------------------------------
    // 1) Kick off async global->LDS staging of this ray's bins row
    //    (first 128 of 129 floats). Issued first; waited on only after CDF build.
    // ---------------------------------------------------------------
    const float* gbins = bins + (size_t)r * (NC + 1);
    {
        // force a wave-uniform SGPR-pair base for the 's' constraint
        unsigned long long ga = (unsigned long long)gbins;
        unsigned glo = __builtin_amdgcn_readfirstlane((unsigned)ga);
        unsigned ghi = __builtin_amdgcn_readfirstlane((unsigned)(ga >> 32));
        unsigned long long ubase = ((unsigned long long)ghi << 32) | glo;

        unsigned ldsBase = (unsigned)(size_t)(&s_bins[w][0]); // LDS byte offset
        #pragma unroll
        for (int j = 0; j < 4; ++j) {
            unsigned idx   = (unsigned)lane + 32u * (unsigned)j; // 0..127
            unsigned laddr = ldsBase + idx * 4u;
            unsigned goff  = idx * 4u;
            asm volatile("global_load_async_to_lds_b32 %0, %1, %2"
                         :: "v"(laddr), "v"(goff), "s"(ubase)
                         : "memory");
        }
    }

    // ---------------------------------------------------------------
    // 2) Per-lane 4 elements: dd = deltas*density, in-lane prefix
    // ---------------------------------------------------------------
    const float4 dv = ((const float4*)(deltas  + (size_t)r * NC))[lane];
    const float4 sv = ((const float4*)(density + (size_t)r * NC))[lane];

    const float dd0 = dv.x * sv.x;
    const float dd1 = dv.y * sv.y;
    const float dd2 = dv.z * sv.z;
    const float dd3 = dv.w * sv.w;

    const float p0 = dd0;
    const float p1 = p0 + dd1;
    const float p2 = p1 + dd2;
    const float p3 = p2 + dd3;       // lane total

    // ---------------------------------------------------------------
    // 3) wave32 inclusive scan of lane totals (5 shuffle steps)
    // ---------------------------------------------------------------
    float t = p3;
    #pragma unroll
    for (int off = 1; off < 32; off <<= 1) {
        float v = __shfl_up(t, (unsigned)off, 32);
        if (lane >= off) t += v;
    }
    const float carry = t - p3;      // exclusive cross-lane prefix of dd

    const float i0 = carry + p0;     // inclusive prefix of dd, elems 4*lane..+3
    const float i1 = carry + p1;
    const float i2 = carry + p2;
    const float i3 = carry + p3;

    // cumsum(weights)[j] telescopes exactly to 1 - exp(-incl_dd[j])
    const float cw0 = 1.0f - __expf(-i0);
    const float cw1 = 1.0f - __expf(-i1);
    const float cw2 = 1.0f - __expf(-i2);
    const float cw3 = 1.0f - __expf(-i3);

    const float wsum    = __shfl(cw3, 31, 32);          // weights_sum
    const float padding = fmaxf(EPSF - wsum, 0.0f);     // relu(EPS - wsum)
    const float wsum2   = wsum + padding;
    const float padPer  = padding * (1.0f / (float)NC); // padding/128

    // cdf[j+1] = min(1, (cumW_j + (j+1)*padding/128) / wsum2)
    const int j0 = 4 * lane;
    const float c0 = fminf(1.0f, (cw0 + (float)(j0 + 1) * padPer) / wsum2);
    const float c1 = fminf(1.0f, (cw1 + (float)(j0 + 2) * padPer) / wsum2);
    const float c2 = fminf(1.0f, (cw2 + (float)(j0 + 3) * padPer) / wsum2);
    const float c3 = fminf(1.0f, (cw3 + (float)(j0 + 4) * padPer) / wsum2);

    float* cdfp = &s_cdf[w][0];
    cdfp[j0 + 1] = c0;
    cdfp[j0 + 2] = c1;
    cdfp[j0 + 3] = c2;
    cdfp[j0 + 4] = c3;
    if (lane == 0) {
        cdfp[0]      = 0.0f;   // prepended zero
        cdfp[NC + 1] = 2.0f;   // sentinel > any u: keeps bisect in range
        s_bins[w][NC] = gbins[NC];   // 129th bin edge (tail of async block)
    }

    // ---------------------------------------------------------------
    // 4) Sync: async bins arrived + same-wave LDS cdf writes ordered
    // ---------------------------------------------------------------
    asm volatile("s_wait_asynccnt 0" ::: "memory");
    __builtin_amdgcn_wave_barrier();   // compiler scheduling fence (intra-wave)

    // ---------------------------------------------------------------
    // 5) searchsorted(cdf, u, right) + lerp on bins (all from LDS)
    // ---------------------------------------------------------------
    const float4 uvec = ((const float4*)(u + (size_t)r * NS))[lane];
    const float uu[4] = {uvec.x, uvec.y, uvec.z, uvec.w};
    const float* binp = &s_bins[w][0];

    float res[4];
    #pragma unroll
    for (int k = 0; k < 4; ++k) {
        const float uk = uu[k];
        int lo = 0, hi = NC + 1;                 // insertion index in [0,129]
        #pragma unroll
        for (int it = 0; it < 8; ++it) {         // 8 steps resolve 130 slots
            const int mid = (lo + hi) >> 1;      // mid <= 129 (sentinel safe)
            const float c = cdfp[mid];
            if (c <= uk) lo = mid + 1; else hi = mid;
        }
        const int below = max(lo - 1, 0);
        const int above = min(lo, NC);

        const float cg0 = cdfp[below];
        const float cg1 = cdfp[above];
        const float bg0 = binp[below];
        const float bg1 = binp[above];

        float denom = cg1 - cg0;
        denom = (denom < EPSF) ? 1.0f : denom;
        const float tt = (uk - cg0) / denom;
        res[k] = bg0 + tt * (bg1 - bg0);
    }

    ((float4*)(out + (size_t)r * NS))[lane] =
        make_float4(res[0], res[1], res[2], res[3]);
}

extern "C" void kernel_launch(void* const* d_in, const int* in_sizes, int n_in,
                              void* d_out, int out_size, void* d_ws, size_t ws_size,
                              hipStream_t stream)
{
    const float* deltas  = (const float*)d_in[0];   // [R,128]
    const float* density = (const float*)d_in[1];   // [R,128,1]
    const float* bins    = (const float*)d_in[2];   // [R,129]
    const float* u       = (const float*)d_in[3];   // [R,128]
    float*       out     = (float*)d_out;           // [R,128]

    const int R = in_sizes[0] / NC;                 // 131072
    const int blocks = (R + WPB - 1) / WPB;         // 16384

    hipLaunchKernelGGL(pdf_sampler_kernel, dim3(blocks), dim3(256), 0, stream,
                       deltas, density, bins, u, out, R);
}